// RetrievalStage_3985729650867
// MI455X (gfx1250) — compile-verified
//
#include <hip/hip_runtime.h>
#include <math.h>

typedef float v2f __attribute__((ext_vector_type(2)));
typedef float v8f __attribute__((ext_vector_type(8)));

constexpr int Dk    = 384;   // embedding dim
constexpr int Bq    = 256;   // queries
constexpr int TOPK  = 100;   // k
constexpr int NT    = 128;   // candidates per block (GEMM)
constexpr int KC    = 16;    // K chunk staged in LDS
constexpr int LDSTR = 20;    // 80B row stride: 16B-aligned for async B128, and
                             // 20*i mod 64 distinct for i=0..15 -> conflict-free
constexpr int NCHUNK= Dk / KC; // 24
constexpr int NBINS = 65536; // 16-bit radix-select histogram

// ---------------------------------------------------------------- query norm
__global__ void qnorm_kernel(const float* __restrict__ q, float* __restrict__ qn) {
  __shared__ float red[128];
  const int row = blockIdx.x, t = threadIdx.x;
  float ss = 0.0f;
  for (int j = t; j < Dk; j += 128) { float v = q[(size_t)row * Dk + j]; ss += v * v; }
  red[t] = ss; __syncthreads();
  for (int s = 64; s > 0; s >>= 1) { if (t < s) red[t] += red[t + s]; __syncthreads(); }
  const float inv = 1.0f / fmaxf(sqrtf(red[0]), 1e-12f);
  for (int j = t; j < Dk; j += 128) qn[(size_t)row * Dk + j] = q[(size_t)row * Dk + j] * inv;
}

// ------------------------- async global->LDS staging (gfx1250 ASYNCcnt path)
// GVS mode: mem_addr = SGPR_U64 base + VGPR_I32 byte offset. VDST VGPR holds
// the LDS byte offset (low 32 bits of the generic shared-pointer address).
__device__ __forceinline__ void async_ld16(unsigned lds_off, unsigned gvoff,
                                           const float* base) {
  asm volatile("global_load_async_to_lds_b128 %0, %1, %2"
               :: "v"(lds_off), "v"(gvoff),
                  "s"((unsigned long long)(size_t)base)
               : "memory");
}
__device__ __forceinline__ void wait_async0() {
  asm volatile("s_wait_asynccnt 0x0" ::: "memory");
}
__device__ __forceinline__ unsigned lds_off_u32(const void* p) {
  return (unsigned)(size_t)p;   // LDS generic address: offset lives in low 32b
}

// ------------------------------------------- fused (bank-norm + GEMM) kernel
// Block: 256 threads (8 wave32). Computes scores for ALL 256 queries x 128
// candidates. Bank rows streamed exactly once via double-buffered async
// copies; per-row sumsq accumulated from the staged tile, normalization
// applied as a post-scale (dot is linear in the bank row).
__global__ __launch_bounds__(256) void gemm_scores_kernel(
    const float* __restrict__ qn, const float* __restrict__ bank,
    float* __restrict__ scores, int N) {
  __shared__ float Qs[2][Bq * LDSTR];
  __shared__ float Bs[2][NT * LDSTR];
  __shared__ float rowsq[NT];
  __shared__ float rinv[NT];

  const int t     = threadIdx.x;
  const int lane  = t & 31;
  const int w     = t >> 5;            // wave 0..7
  const int wq    = w & 3;             // 4 groups over queries (64 each)
  const int wn    = w >> 2;            // 2 groups over candidates (64 each)
  const int m     = lane & 15;
  const int khalf = (lane >> 4) << 1;  // 0 or 2 (ISA f32 A/B fragment layout)
  const int nbase = blockIdx.x * NT;
  const int brow  = t >> 1;            // bank row this thread stages (0..127)
  const int bseg  = t & 1;             // which 8-wide K segment of the row
  // clamp OOB rows so every wave issues a uniform async count per chunk
  const int growc = (nbase + brow < N) ? (nbase + brow) : (N - 1);

  const unsigned q_voff0 = (unsigned)t * (unsigned)(Dk * 4);
  const unsigned b_voff0 = (unsigned)growc * (unsigned)(Dk * 4) + bseg * 32u;
  unsigned qs_lds[2], bs_lds[2];
  qs_lds[0] = lds_off_u32(&Qs[0][t * LDSTR]);
  qs_lds[1] = lds_off_u32(&Qs[1][t * LDSTR]);
  bs_lds[0] = lds_off_u32(&Bs[0][brow * LDSTR + bseg * 8]);
  bs_lds[1] = lds_off_u32(&Bs[1][brow * LDSTR + bseg * 8]);

  v8f acc[4][4] = {};
  float ssq = 0.0f;

  // prefetch chunk 0 into buffer 0 (6 async B128 per thread-slot)
  #pragma unroll
  for (int i = 0; i < 4; ++i) async_ld16(qs_lds[0] + i * 16, q_voff0 + i * 16, qn);
  #pragma unroll
  for (int j = 0; j < 2; ++j) async_ld16(bs_lds[0] + j * 16, b_voff0 + j * 16, bank);

  int buf = 0;
  #pragma unroll 1
  for (int c = 0; c < NCHUNK; ++c) {
    wait_async0();        // chunk c landed in LDS (per-wave)
    __syncthreads();      // visible to all waves; all done reading other buf
    if (c + 1 < NCHUNK) { // prefetch chunk c+1 into the other buffer
      const unsigned kb4 = (unsigned)(c + 1) * (unsigned)(KC * 4);
      const int nb = buf ^ 1;
      #pragma unroll
      for (int i = 0; i < 4; ++i)
        async_ld16(qs_lds[nb] + i * 16, q_voff0 + kb4 + i * 16, qn);
      #pragma unroll
      for (int j = 0; j < 2; ++j)
        async_ld16(bs_lds[nb] + j * 16, b_voff0 + kb4 + j * 16, bank);
    }

    // per-row sum of squares from the freshly staged bank tile (16B aligned)
    {
      const float4* p =
          reinterpret_cast<const float4*>(&Bs[buf][brow * LDSTR + bseg * 8]);
      const float4 x0 = p[0], x1 = p[1];
      ssq += x0.x*x0.x + x0.y*x0.y + x0.z*x0.z + x0.w*x0.w
           + x1.x*x1.x + x1.y*x1.y + x1.z*x1.z + x1.w*x1.w;
    }

    // 4 K-steps of V_WMMA_F32_16X16X4_F32 over the staged 16-wide chunk
    const float* Qb = &Qs[buf][0];
    const float* Bb = &Bs[buf][0];
    #pragma unroll
    for (int kk = 0; kk < KC; kk += 4) {
      v2f af[4], bf[4];
      #pragma unroll
      for (int i = 0; i < 4; ++i) {
        const int qidx = (wq * 64 + i * 16 + m) * LDSTR + kk + khalf;
        af[i].x = Qb[qidx]; af[i].y = Qb[qidx + 1];
        const int nidx = (wn * 64 + i * 16 + m) * LDSTR + kk + khalf;
        bf[i].x = Bb[nidx]; bf[i].y = Bb[nidx + 1];
      }
      #pragma unroll
      for (int qi = 0; qi < 4; ++qi)
        #pragma unroll
        for (int ni = 0; ni < 4; ++ni)
          acc[qi][ni] = __builtin_amdgcn_wmma_f32_16x16x4_f32(
              false, af[qi], false, bf[ni], (short)0, acc[qi][ni], false, false);
    }
    buf ^= 1;
  }

  // combine per-row sumsq halves -> inverse norms
  __syncthreads();
  if (bseg == 0) rowsq[brow] = ssq;
  __syncthreads();
  if (bseg == 1) rowsq[brow] += ssq;
  __syncthreads();
  if (t < NT) rinv[t] = 1.0f / fmaxf(sqrtf(rowsq[t]), 1e-12f);
  __syncthreads();

  // post-scale by bank-row inverse norm, store (coalesced along n)
  #pragma unroll
  for (int ni = 0; ni < 4; ++ni) {
    const int nloc = wn * 64 + ni * 16 + (lane & 15);
    const int ng   = nbase + nloc;
    if (ng >= N) continue;
    const float s = rinv[nloc];
    #pragma unroll
    for (int qi = 0; qi < 4; ++qi) {
      const int qb = wq * 64 + qi * 16 + ((lane >> 4) << 3);
      #pragma unroll
      for (int r = 0; r < 8; ++r)
        scores[(size_t)(qb + r) * N + ng] = acc[qi][ni][r] * s;
    }
  }
}

// ----------------------------------------------------- per-query top-k select
__device__ __forceinline__ unsigned fkey(float f) {
  unsigned u = __float_as_uint(f);
  return (u & 0x80000000u) ? ~u : (u | 0x80000000u);  // order-preserving
}

__global__ __launch_bounds__(256) void topk_kernel(
    const float* __restrict__ scores, float* __restrict__ outS,
    int* __restrict__ outI, int N) {
  extern __shared__ unsigned hist[];       // 65536 bins = 256 KB (CDNA5 LDS)
  __shared__ float topS[128];
  __shared__ int   topI[128];
  __shared__ unsigned part[256];
  __shared__ int sh_thr; __shared__ unsigned sh_cgt;
  __shared__ unsigned cntGT, cntEQ;

  const int q = blockIdx.x, t = threadIdx.x;
  const float* row = scores + (size_t)q * N;

  for (int b = t; b < NBINS; b += 256) hist[b] = 0;
  if (t == 0) { cntGT = 0; cntEQ = 0; }
  __syncthreads();

  // pass 1: histogram of top-16 key bits
  for (int i = t; i < N; i += 256) atomicAdd(&hist[fkey(row[i]) >> 16], 1u);
  __syncthreads();

  // find threshold bin (descending cumulative >= k), two-level scan
  unsigned ps = 0;
  for (int b = 0; b < 256; ++b) ps += hist[t * 256 + b];
  part[t] = ps; __syncthreads();
  if (t == 0) {
    unsigned acc = 0; int seg = 255;
    for (; seg > 0; --seg) { if (acc + part[seg] >= (unsigned)TOPK) break; acc += part[seg]; }
    int b = seg * 256 + 255; unsigned a2 = acc;
    for (; b > seg * 256; --b) { if (a2 + hist[b] >= (unsigned)TOPK) break; a2 += hist[b]; }
    sh_thr = b; sh_cgt = a2;   // a2 = count strictly above bin b
  }
  __syncthreads();
  const int thr = sh_thr; const unsigned cgt = sh_cgt;

  // pass 2: gather winners (all > thr guaranteed; == thr fills the remainder)
  for (int i = t; i < N; i += 256) {
    const float s = row[i];
    const unsigned b = fkey(s) >> 16;
    if ((int)b > thr) {
      unsigned p = atomicAdd(&cntGT, 1u);
      topS[p] = s; topI[p] = i;
    } else if ((int)b == thr) {
      unsigned p = cgt + atomicAdd(&cntEQ, 1u);
      if (p < (unsigned)TOPK) { topS[p] = s; topI[p] = i; }
    }
  }
  __syncthreads();
  if (t >= TOPK && t < 128) { topS[t] = -__builtin_inff(); topI[t] = 0x7fffffff; }
  __syncthreads();

  // bitonic sort 128 entries, descending score (ties: ascending index)
  for (int ksz = 2; ksz <= 128; ksz <<= 1) {
    for (int j = ksz >> 1; j > 0; j >>= 1) {
      if (t < 128) {
        const int ixj = t ^ j;
        if (ixj > t) {
          float s1 = topS[t], s2 = topS[ixj];
          int   i1 = topI[t], i2 = topI[ixj];
          const bool firstBetter = (s1 > s2) || (s1 == s2 && i1 < i2);
          const bool up = ((t & ksz) == 0);
          if (up ? !firstBetter : firstBetter) {
            topS[t] = s2; topS[ixj] = s1; topI[t] = i2; topI[ixj] = i1;
          }
        }
      }
      __syncthreads();
    }
  }
  if (t < TOPK) {
    outS[(size_t)q * TOPK + t] = topS[t];
    outI[(size_t)q * TOPK + t] = topI[t];
  }
}

// ----------------------------------------------------------------- launcher
extern "C" void kernel_launch(void* const* d_in, const int* in_sizes, int n_in,
                              void* d_out, int out_size, void* d_ws, size_t ws_size,
                              hipStream_t stream) {
  (void)n_in; (void)out_size; (void)ws_size;
  const float* q    = (const float*)d_in[0];
  const float* bank = (const float*)d_in[1];
  const int B = in_sizes[0] / Dk;   // 256
  const int N = in_sizes[1] / Dk;   // 500000

  // workspace: [ q_norm (B*Dk f32) | scores (B*N f32, ~512 MB) ]
  float* qn = (float*)d_ws;
  const size_t off = (((size_t)B * Dk * sizeof(float)) + 255) & ~(size_t)255;
  float* scores = (float*)((char*)d_ws + off);

  float* outS = (float*)d_out;
  int*   outI = (int*)((float*)d_out + (size_t)B * TOPK);

  qnorm_kernel<<<B, 128, 0, stream>>>(q, qn);

  const int grid2 = (N + NT - 1) / NT;
  gemm_scores_kernel<<<grid2, 256, 0, stream>>>(qn, bank, scores, N);

  (void)hipFuncSetAttribute((const void*)topk_kernel,
                            hipFuncAttributeMaxDynamicSharedMemorySize,
                            NBINS * (int)sizeof(unsigned));
  topk_kernel<<<B, 256, NBINS * sizeof(unsigned), stream>>>(scores, outS, outI, N);
}